// TensorSquare_71554155151992
// MI455X (gfx1250) — compile-verified
//
#include <hip/hip_runtime.h>

typedef float v4f __attribute__((ext_vector_type(4)));
typedef float v2f __attribute__((ext_vector_type(2)));
typedef int   v4i __attribute__((ext_vector_type(4)));

// Pointer types matching the async builtin's signature:
//   __builtin_amdgcn_global_load_async_to_lds_b128(int4 AS1*, int4 AS3*, Imm, Imm)
typedef __attribute__((address_space(1))) v4i* ts_g4;
typedef __attribute__((address_space(3))) v4i* ts_l4;

#if defined(__gfx1250__) && __has_builtin(__builtin_amdgcn_global_load_async_to_lds_b128)
#define TS_ASYNC_LDS 1
#else
#define TS_ASYNC_LDS 0
#endif

__device__ __forceinline__ void ts_wait_async() {
#if __has_builtin(__builtin_amdgcn_s_wait_asynccnt)
  __builtin_amdgcn_s_wait_asynccnt(0);
#else
  asm volatile("s_wait_asynccnt 0x0" ::: "memory");
#endif
  asm volatile("" ::: "memory");  // keep LDS reads below the wait
}

__launch_bounds__(256)
__global__ void tensor_square_kernel(const float* __restrict__ x,
                                     float* __restrict__ out, int N) {
  constexpr float SQRT2     = 1.41421356237309504880f;
  constexpr float INV_SQRT3 = 0.57735026918962576451f;
  constexpr float INV_SQRT6 = 0.40824829046386301637f;
  constexpr float INV_SQRT2 = 0.70710678118654752440f;
  // triu_indices(4): 10 pairs; OFF = strict upper (6 pairs)
  constexpr int IU[10] = {0,0,0,0,1,1,1,2,2,3};
  constexpr int IV[10] = {0,1,2,3,1,2,3,2,3,3};
  constexpr int OU[6]  = {0,0,0,1,1,2};
  constexpr int OV[6]  = {1,2,3,2,3,3};

  const int lane      = threadIdx.x & 31;
  const int waveInBlk = threadIdx.x >> 5;
  const int gwave     = (int)((blockIdx.x * blockDim.x + threadIdx.x) >> 5);
  const int r0        = gwave * 2;            // wave handles rows r0, r0+1
  if (r0 >= N) return;

  __shared__ __align__(16) float smem[8][512];   // 2 rows * 256 floats per wave

  // ---- Stage 2 input rows (512 floats) into this wave's LDS buffer ----
  // Tile layout: byte (k*512 + lane*16), k = 0..3.  Async loads add the
  // immediate INST_OFFSET to BOTH the global and the LDS address, so all
  // four transfers share one address pair and differ only in the immediate.
  const float* gbase = x + (size_t)r0 * 256 + lane * 4;      // lane*16 bytes
  float*       lbase = &smem[waveInBlk][lane * 4];
  const bool fullTile = (r0 + 2 <= N);                       // uniform per wave

#if TS_ASYNC_LDS
  if (fullTile) {
    __builtin_amdgcn_global_load_async_to_lds_b128((ts_g4)gbase, (ts_l4)lbase, 0,    0);
    __builtin_amdgcn_global_load_async_to_lds_b128((ts_g4)gbase, (ts_l4)lbase, 512,  0);
    __builtin_amdgcn_global_load_async_to_lds_b128((ts_g4)gbase, (ts_l4)lbase, 1024, 0);
    __builtin_amdgcn_global_load_async_to_lds_b128((ts_g4)gbase, (ts_l4)lbase, 1536, 0);
  } else {
    // Tail (N odd, last wave): only row r0 valid -> first 256 floats (k=0,1).
    __builtin_amdgcn_global_load_async_to_lds_b128((ts_g4)gbase, (ts_l4)lbase, 0,   0);
    __builtin_amdgcn_global_load_async_to_lds_b128((ts_g4)gbase, (ts_l4)lbase, 512, 0);
  }
  ts_wait_async();
#else
  const int kmax = fullTile ? 4 : 2;
  for (int k = 0; k < kmax; ++k)
    *(v4f*)(lbase + k * 128) = *(const v4f*)(gbase + k * 128);
#endif

  const int r = r0 + (lane >> 4);
  if (r >= N) return;
  const int f = lane & 15;
  const float* row = &smem[waveInBlk][(lane >> 4) * 256];

  // ---- Gather operands from LDS (all 16B aligned -> ds_load_b128) ----
  float s[4];
  {
    v4f t = *(const v4f*)(row + f * 4);
    s[0] = t.x; s[1] = t.y; s[2] = t.z; s[3] = t.w;
  }
  float v[4][3];
  {
    float tmp[12];
    *(v4f*)&tmp[0] = *(const v4f*)(row + 64 + f * 12);
    *(v4f*)&tmp[4] = *(const v4f*)(row + 64 + f * 12 + 4);
    *(v4f*)&tmp[8] = *(const v4f*)(row + 64 + f * 12 + 8);
    #pragma unroll
    for (int m = 0; m < 4; ++m)
      #pragma unroll
      for (int c = 0; c < 3; ++c)
        v[m][c] = tmp[m * 3 + c];
  }

  // ---- Compute the 136 outputs of this (row, f_ax) ----
  float o0[20];   // ss[10] ++ vv0[10]
  #pragma unroll
  for (int p = 0; p < 10; ++p) {
    const float fp = (IU[p] < IV[p]) ? SQRT2 : 1.0f;
    o0[p] = s[IU[p]] * s[IV[p]] * fp;
    const float d = v[IU[p]][0] * v[IV[p]][0] +
                    v[IU[p]][1] * v[IV[p]][1] +
                    v[IU[p]][2] * v[IV[p]][2];
    o0[10 + p] = d * (fp * INV_SQRT3);
  }

  float o1[18];   // vv1: cross products of strict-upper pairs
  #pragma unroll
  for (int q = 0; q < 6; ++q) {
    const float ax = v[OU[q]][0], ay = v[OU[q]][1], az = v[OU[q]][2];
    const float bx = v[OV[q]][0], by = v[OV[q]][1], bz = v[OV[q]][2];
    o1[q * 3 + 0] = ay * bz - az * by;
    o1[q * 3 + 1] = az * bx - ax * bz;
    o1[q * 3 + 2] = ax * by - ay * bx;
  }

  float oo[48];   // sv: s[m] * v[mp][c]
  #pragma unroll
  for (int m = 0; m < 4; ++m)
    #pragma unroll
    for (int mp = 0; mp < 4; ++mp)
      #pragma unroll
      for (int c = 0; c < 3; ++c)
        oo[(m * 4 + mp) * 3 + c] = s[m] * v[mp][c];

  float o2[50];   // vv2: symmetric-traceless 5-vector per pair
  #pragma unroll
  for (int p = 0; p < 10; ++p) {
    const float fp = (IU[p] < IV[p]) ? SQRT2 : 1.0f;
    const float ax = v[IU[p]][0], ay = v[IU[p]][1], az = v[IU[p]][2];
    const float bx = v[IV[p]][0], by = v[IV[p]][1], bz = v[IV[p]][2];
    o2[p * 5 + 0] = (ax * by + ay * bx) * INV_SQRT2 * fp;
    o2[p * 5 + 1] = (ay * bz + az * by) * INV_SQRT2 * fp;
    o2[p * 5 + 2] = (2.0f * az * bz - ax * bx - ay * by) * INV_SQRT6 * fp;
    o2[p * 5 + 3] = (ax * bz + az * bx) * INV_SQRT2 * fp;
    o2[p * 5 + 4] = (ax * bx - ay * by) * INV_SQRT2 * fp;
  }

  // ---- Streaming (non-temporal) coalesced stores ----
  float* base = out + (size_t)r * 2176;

  float* p0 = base + f * 20;            // byte off f*80  -> 16B aligned
  #pragma unroll
  for (int i = 0; i < 5; ++i)
    __builtin_nontemporal_store(*(v4f*)&o0[i * 4], (v4f*)(p0 + i * 4));

  float* p1 = base + 320 + f * 18;      // byte off 1280+72f -> 8B aligned
  #pragma unroll
  for (int i = 0; i < 9; ++i)
    __builtin_nontemporal_store(*(v2f*)&o1[i * 2], (v2f*)(p1 + i * 2));

  float* p2 = base + 608 + f * 48;      // byte off 2432+192f -> 16B aligned
  #pragma unroll
  for (int i = 0; i < 12; ++i)
    __builtin_nontemporal_store(*(v4f*)&oo[i * 4], (v4f*)(p2 + i * 4));

  float* p3 = base + 1376 + f * 50;     // byte off 5504+200f -> 8B aligned
  #pragma unroll
  for (int i = 0; i < 25; ++i)
    __builtin_nontemporal_store(*(v2f*)&o2[i * 2], (v2f*)(p3 + i * 2));
}

extern "C" void kernel_launch(void* const* d_in, const int* in_sizes, int n_in,
                              void* d_out, int out_size, void* d_ws, size_t ws_size,
                              hipStream_t stream) {
  (void)n_in; (void)out_size; (void)d_ws; (void)ws_size;
  const float* x = (const float*)d_in[0];
  float* out = (float*)d_out;
  const int N = in_sizes[0] / 256;            // rows
  const int waves = (N + 1) / 2;              // one wave per 2 rows
  const int blocks = (waves + 7) / 8;         // 8 waves (256 threads) per block
  tensor_square_kernel<<<blocks, 256, 0, stream>>>(x, out, N);
}